// EdgeNetwork_26749056319850
// MI455X (gfx1250) — compile-verified
//
#include <hip/hip_runtime.h>
#include <hip/hip_bf16.h>
#include <math.h>

typedef __attribute__((ext_vector_type(2))) float v2f;
typedef __attribute__((ext_vector_type(8))) float v8f;
typedef __attribute__((ext_vector_type(4))) float f4;

#define B_N 1024
#define D_F 128
#define TI  64
#define TJ  64
#define LDP 68   // LDS row pad: 68*4B -> bank stride 4, keeps 16B alignment for b128

// ---------------------------------------------------------------------------
// Kernel 1: front GEMMs with fp32 WMMA (V_WMMA_F32_16X16X4_F32).
//   H1 = F @ W1[0:128, :]   + b1        (bias folded here)
//   H2 = F @ W1[128:256, :]
// One wave (32 threads) per 16x16 output tile; K=128 swept in 32 steps of 4.
// A layout (16x4 f32): lanes 0-15 rows M, VGPR{0,1}=K{0,1}; lanes 16-31 K{2,3}.
// B layout (4x16 f32): mirrored with N striped across lanes.
// C layout (16x16 f32, 8 VGPRs): VGPR v -> M = v + 8*(lane>=16), N = lane&15.
// ---------------------------------------------------------------------------
__global__ __launch_bounds__(32) void edge_front_gemm(
    const float* __restrict__ F, const float* __restrict__ W1,
    const float* __restrict__ b1, float* __restrict__ H1,
    float* __restrict__ H2)
{
  const int bid   = blockIdx.x;     // 0..1023
  const int which = bid >> 9;       // 0 -> H1, 1 -> H2
  const int t     = bid & 511;
  const int i0    = (t >> 3) << 4;  // row tile base (0..1008)
  const int d0    = (t & 7) << 4;   // col tile base (0..112)

  const int lane = threadIdx.x;
  const int m    = lane & 15;
  const int half = lane >> 4;       // 0: K{0,1}, 1: K{2,3}
  const int kb   = half * 2;

  const float* __restrict__ Frow  = F + (i0 + m) * D_F;
  const float* __restrict__ Wcol  = W1 + which * D_F * D_F + d0 + m;

  v8f c = {};
#pragma unroll 4
  for (int k = 0; k < D_F; k += 4) {
    v2f a, b;
    a.x = Frow[k + kb];
    a.y = Frow[k + kb + 1];
    b.x = Wcol[(k + kb) * D_F];
    b.y = Wcol[(k + kb + 1) * D_F];
    c = __builtin_amdgcn_wmma_f32_16x16x4_f32(
        /*neg_a=*/false, a, /*neg_b=*/false, b,
        /*c_mod=*/(short)0, c, /*reuse_a=*/false, /*reuse_b=*/false);
  }

  float* __restrict__ H = which ? H2 : H1;
  const int   col  = d0 + m;
  const float bias = which ? 0.0f : b1[col];
#pragma unroll
  for (int v = 0; v < 8; ++v) {
    const int row = i0 + v + half * 8;
    H[row * D_F + col] = c[v] + bias;
  }
}

// ---------------------------------------------------------------------------
// Kernel 2: pairwise score. 64x64 output tile per 256-thread block.
// LDS holds Hi/Hj tiles d-major ([128][64+pad]) so the inner loop is two
// ds_load_b128 per d per thread feeding a 4x4 register micro-tile:
//   acc[a][b] += w2[d] * max(hi[a] + hj[b], 0)
// Epilogue: sigmoid + (off-diagonal & same-label) mask, float4 stores.
// ---------------------------------------------------------------------------
__global__ __launch_bounds__(256) void edge_score(
    const float* __restrict__ H1, const float* __restrict__ H2,
    const float* __restrict__ w2, const float* __restrict__ b2p,
    const int*   __restrict__ labels, float* __restrict__ out)
{
  __shared__ float his[D_F][LDP];
  __shared__ float hjs[D_F][LDP];
  __shared__ float w2s[D_F];
  __shared__ int   li_s[TI];
  __shared__ int   lj_s[TJ];

  const int i0  = blockIdx.y * TI;
  const int j0  = blockIdx.x * TJ;
  const int tid = threadIdx.x;

  __builtin_prefetch(&H1[i0 * D_F], 0, 0);  // global_prefetch_b8
  __builtin_prefetch(&H2[j0 * D_F], 0, 0);

  // Fill LDS (transpose: coalesced global reads, 2-way-conflict LDS writes).
  for (int e = tid; e < TI * D_F; e += 256) {
    const int ii = e >> 7;     // row within tile
    const int d  = e & 127;
    his[d][ii] = H1[(i0 + ii) * D_F + d];
    hjs[d][ii] = H2[(j0 + ii) * D_F + d];
  }
  if (tid < D_F) w2s[tid] = w2[tid];
  if (tid < TI)  { li_s[tid] = labels[i0 + tid]; lj_s[tid] = labels[j0 + tid]; }
  __syncthreads();

  const int tx = tid & 15;        // j micro-tile index
  const int ty = tid >> 4;        // i micro-tile index

  float acc[4][4] = {};
#pragma unroll 2
  for (int d = 0; d < D_F; ++d) {
    const f4 hi = *(const f4*)&his[d][ty * 4];
    const f4 hj = *(const f4*)&hjs[d][tx * 4];
    const float w = w2s[d];
#pragma unroll
    for (int a = 0; a < 4; ++a)
#pragma unroll
      for (int b = 0; b < 4; ++b)
        acc[a][b] = fmaf(w, fmaxf(hi[a] + hj[b], 0.0f), acc[a][b]);
  }

  const float b2 = *b2p;
#pragma unroll
  for (int a = 0; a < 4; ++a) {
    const int i  = i0 + ty * 4 + a;
    const int li = li_s[ty * 4 + a];
    f4 r;
#pragma unroll
    for (int b = 0; b < 4; ++b) {
      const int j = j0 + tx * 4 + b;
      const float s = 1.0f / (1.0f + __expf(-(acc[a][b] + b2)));
      const bool keep = (i != j) && (li == lj_s[tx * 4 + b]);
      r[b] = keep ? s : 0.0f;
    }
    *(f4*)&out[i * B_N + j0 + tx * 4] = r;
  }
}

// ---------------------------------------------------------------------------
extern "C" void kernel_launch(void* const* d_in, const int* in_sizes, int n_in,
                              void* d_out, int out_size, void* d_ws, size_t ws_size,
                              hipStream_t stream) {
  const float* F      = (const float*)d_in[0];   // [1024,128]
  const int*   labels = (const int*)  d_in[1];   // [1024] (JAX default -> int32)
  const float* W1     = (const float*)d_in[2];   // [256,128]
  const float* b1     = (const float*)d_in[3];   // [128]
  const float* w2     = (const float*)d_in[4];   // [128]
  const float* b2     = (const float*)d_in[5];   // [1]
  float*       out    = (float*)d_out;           // [1024,1024]

  float* H1 = (float*)d_ws;                      // [1024,128]
  float* H2 = H1 + B_N * D_F;                    // [1024,128]

  // 2 matrices * (64 row-tiles * 8 col-tiles) = 1024 waves of WMMA GEMM.
  edge_front_gemm<<<1024, 32, 0, stream>>>(F, W1, b1, H1, H2);

  dim3 grid(B_N / TJ, B_N / TI);                 // 16 x 16 tiles
  edge_score<<<grid, 256, 0, stream>>>(H1, H2, w2, b2, labels, out);
}